// Baseline_PartPP_Single_27324581937301
// MI455X (gfx1250) — compile-verified
//
#include <hip/hip_runtime.h>
#include <hip/hip_bf16.h>

// MI455X (gfx1250) implementation.
// Memory-bound op: 251.7 MB of x read once -> ~11.6 us floor @ 23.3 TB/s.
// Segment-sum done as a one-hot GEMM with V_WMMA_F32_16X16X4_F32 (exact f32),
// segment-max via LDS-compacted per-part pixel lists (1 LDS read / element).

typedef float v2f __attribute__((ext_vector_type(2)));
typedef float v8f __attribute__((ext_vector_type(8)));

#define PARTS   16
#define HW      256
#define CHANS   256
#define SDIM    30
#define NDIM    32
#define CTILE   16
#define NTILES  (CHANS / CTILE)   // 16
#define TPAD    17                // padded row stride (coprime with 64 banks)

__global__ __launch_bounds__(256)
void partpp_wmma_kernel(const float* __restrict__ x,
                        const long long* __restrict__ labels64,
                        float* __restrict__ out)
{
    __shared__ float          tile[HW * TPAD];     // 17408 B, feat [px][ch]
    __shared__ float          red[8 * 256];        //  8192 B, per-wave WMMA partials
    __shared__ int            lab[HW];             //  1024 B
    __shared__ unsigned       cnt[PARTS];          //    64 B
    __shared__ unsigned short list[PARTS * HW];    //  8192 B

    const int tid   = threadIdx.x;        // 256 threads = 8 wave32
    const int wave  = tid >> 5;
    const int lane  = tid & 31;
    const int frame = blockIdx.x;         // 0 .. ns-1 (960)
    const int nIdx  = frame / SDIM;
    const int sIdx  = frame % SDIM;

    // ---------- phase 0: labels -> LDS, per-part pixel-list compaction ----------
    if (tid < PARTS) cnt[tid] = 0u;
    lab[tid] = (int)labels64[(size_t)frame * HW + tid];
    __syncthreads();
    {
        const int l = lab[tid];
        const unsigned slot = atomicAdd(&cnt[l], 1u);
        list[l * HW + slot] = (unsigned short)tid;
    }
    __syncthreads();

    // ---------- one-hot A fragments for V_WMMA_F32_16X16X4_F32 ----------
    // A is 16x4 (part x pixel). Lane layout: lane&15 = row M (part),
    // VGPR0 holds K = ko, VGPR1 holds K = ko+1, with ko = (lane>>4)*2.
    // Constant across all 16 channel tiles -> precompute all 8 K-steps.
    const int m  = lane & 15;
    const int ko = (lane >> 4) << 1;
    v2f afrag[8];
#pragma unroll
    for (int kk = 0; kk < 8; ++kk) {
        const int k = (wave << 5) + (kk << 2) + ko;   // this wave's 32-pixel K chunk
        afrag[kk].x = (lab[k]     == m) ? 1.0f : 0.0f;
        afrag[kk].y = (lab[k + 1] == m) ? 1.0f : 0.0f;
    }

    // finalize-phase ownership: one (channel-in-tile, part) cell per thread
    const int chl = tid & 15;
    const int pl  = tid >> 4;
    const unsigned cnt_p = cnt[pl];

    // x index: ((n*C + c)*S + s)*HW + px
    const size_t xFrameBase = ((size_t)nIdx * CHANS * SDIM + (size_t)sIdx) * HW;

    for (int t = 0; t < NTILES; ++t) {
        const int c0 = t * CTILE;
        __syncthreads();   // previous iteration's readers of tile/red are done

        // ---- stage 16 channels x 256 px (coalesced rows, LDS transpose) ----
#pragma unroll
        for (int i = 0; i < CTILE; ++i) {
            tile[tid * TPAD + i] =
                x[xFrameBase + (size_t)(c0 + i) * (SDIM * HW) + tid];
        }
        __syncthreads();

        // ---- WMMA one-hot GEMM: partial 16x16 (part x channel) sums ----
        // B is 4x16 (pixel x channel): lane&15 = column N, VGPR0 = row ko,
        // VGPR1 = row ko+1 (mirror of A layout).
        v8f acc = {};
#pragma unroll
        for (int kk = 0; kk < 8; ++kk) {
            const int kb = (wave << 5) + (kk << 2) + ko;
            v2f b;
            b.x = tile[kb * TPAD + (lane & 15)];
            b.y = tile[(kb + 1) * TPAD + (lane & 15)];
            acc = __builtin_amdgcn_wmma_f32_16x16x4_f32(
                      /*neg_a=*/false, afrag[kk], /*neg_b=*/false, b,
                      /*c_mod=*/(short)0, acc, /*reuse_a=*/false, /*reuse_b=*/false);
        }
        // C/D layout: VGPR r, lanes0-15 -> M=r, lanes16-31 -> M=r+8, N=lane&15
#pragma unroll
        for (int r = 0; r < 8; ++r)
            red[wave * 256 + r * 32 + lane] = acc[r];
        __syncthreads();

        // ---- finalize: cross-wave sum reduce + segmented max + output ----
        float sum = 0.0f;
        {
            const int idx = (pl & 7) * 32 + ((pl >> 3) << 4) + chl;
#pragma unroll
            for (int w = 0; w < 8; ++w) sum += red[w * 256 + idx];
        }
        float mx = -3.0e38f;
        for (unsigned i = 0; i < cnt_p; ++i) {
            const int px = (int)list[pl * HW + i];
            mx = fmaxf(mx, tile[px * TPAD + chl]);
        }
        const float mean  = sum / (float)((cnt_p > 0u) ? cnt_p : 1u);
        const float mpart = (cnt_p > 0u) ? fmaxf(mx, -100.0f) : 0.0f;
        // out: (n, c, s, P)
        out[(((size_t)nIdx * CHANS + (size_t)(c0 + chl)) * SDIM + sIdx) * PARTS + pl]
            = mean + mpart;
    }
}

extern "C" void kernel_launch(void* const* d_in, const int* in_sizes, int n_in,
                              void* d_out, int out_size, void* d_ws, size_t ws_size,
                              hipStream_t stream) {
    (void)in_sizes; (void)n_in; (void)d_ws; (void)ws_size; (void)out_size;
    const float*     x      = (const float*)d_in[0];
    const long long* labels = (const long long*)d_in[1];
    float*           out    = (float*)d_out;

    const int nframes = NDIM * SDIM;  // 960 workgroups, one per (n, s) frame
    partpp_wmma_kernel<<<nframes, 256, 0, stream>>>(x, labels, out);
}